// GCN_3075196584310
// MI455X (gfx1250) — compile-verified
//
#include <hip/hip_runtime.h>

// GCN: H1 = relu(adj @ (x@W1) + b1); out = relu(adj @ (H1@W2) + b2)
// adj (10000^2 fp32, 400MB) streams from HBM twice -> ~34us floor @23.3TB/s.
// fp32 WMMA 16x16x4 keeps reference precision and sits within ~2x of that
// roofline, so no low-precision conversion.
//
// Data movement:
//  - A tiles (the 400MB adj stream) are staged into LDS with coalesced
//    GLOBAL_LOAD_ASYNC_TO_LDS_B128 (ASYNCcnt-tracked), 4-deep buffered per
//    wave (each wave stages only its own 16 rows -> no barriers, only
//    s_wait_asynccnt ping-pong; prefetch distance 3 chunks covers HBM
//    latency). LDS row stride 36 floats = conflict-free ds_load_b64 reads.
//  - B is pre-packed into paired-K layout so each 4x16 B fragment is a single
//    coalesced global_load_b64 (L2-resident), loaded into distinct registers
//    ahead of the WMMA burst so loads overlap compute (s_clause batches with
//    partial s_wait_loadcnt).

typedef __attribute__((ext_vector_type(2))) float v2f;
typedef __attribute__((ext_vector_type(8))) float v8f;

#define WAVES_PER_BLOCK 8
#define ROWS_PER_BLOCK 128        // 8 waves x 16 rows
#define KC 16                     // K-chunk staged per step (16 floats = 64B/row)
#define LDS_STRIDE 36             // floats per staged row: 16 data + 20 pad
                                  // -> 144B (16B aligned), bank-conflict-free
#define NBUF 4                    // staging depth (prefetch distance 3)

__device__ __forceinline__ void async_copy_b128(const float* gsrc, float* ldst) {
  auto l3 = (__attribute__((address_space(3))) float*)ldst;
  asm volatile("global_load_async_to_lds_b128 %0, %1, off"
               :: "v"(l3), "v"(gsrc)
               : "memory");
}

template <int NT>
__device__ __forceinline__ void compute_chunk(const float* sbase, const float* __restrict__ Bp,
                                              int kg0, int ldb, int l16, int half,
                                              v8f (&acc)[NT]) {
#pragma unroll
  for (int ks = 0; ks < KC / 4; ++ks) {
    // A 16x4 fragment from LDS (ds_load_2addr_b64, conflict-free)
    v2f a = *(const v2f*)(sbase + ks * 4);
    // B 4x16 fragments from packed-pair layout: one b64 per tile, coalesced.
    const float* bb = Bp + ((size_t)((kg0 + ks * 4) / 2 + half) * (size_t)ldb + (size_t)l16) * 2;
    v2f bf[NT];
#pragma unroll
    for (int t = 0; t < NT; ++t) bf[t] = *(const v2f*)(bb + t * 32);
#pragma unroll
    for (int t = 0; t < NT; ++t)
      acc[t] = __builtin_amdgcn_wmma_f32_16x16x4_f32(
          false, a, false, bf[t], (short)0, acc[t], false, false);
  }
}

// C[M x N] = A[M x K] @ Bpacked[K x N] (+bias, +relu).  NT = N/16.
// K must be a multiple of KC and K/KC > NBUF-1 (holds: 512, 10000, 256).
// PACKOUT: write C in paired-K packed layout (when C is the next GEMM's B).
template <int NT, bool RELU, bool HASBIAS, bool PACKOUT>
__global__ __launch_bounds__(256) void gcn_gemm(
    const float* __restrict__ A, const float* __restrict__ Bp,
    const float* __restrict__ bias, float* __restrict__ C,
    int M, int K, int lda, int ldb) {
  __shared__ float sA[NBUF][ROWS_PER_BLOCK * LDS_STRIDE];

  const int lane = threadIdx.x & 31;
  const int wave = threadIdx.x >> 5;
  const int half = lane >> 4;   // 0: lanes 0-15, 1: lanes 16-31
  const int l16  = lane & 15;
  const int kk   = half * 2;

  const int m0 = blockIdx.x * ROWS_PER_BLOCK + wave * 16;

  // --- staging addresses: this wave stages its own 16 rows x KC floats.
  // 64 16-byte segments per chunk; lane covers segment 'lane' and 'lane+32'.
  const int srow = lane >> 2;          // 0..7
  const int p0   = (lane & 3) * 4;     // float offset within row
  int r0 = m0 + srow;      if (r0 >= M) r0 = M - 1;
  int r1 = m0 + srow + 8;  if (r1 >= M) r1 = M - 1;
  const float* g0 = A + (size_t)r0 * (size_t)lda + (size_t)p0;
  const float* g1 = A + (size_t)r1 * (size_t)lda + (size_t)p0;
  float* l0[NBUF];
  float* l1[NBUF];
#pragma unroll
  for (int b = 0; b < NBUF; ++b) {
    l0[b] = &sA[b][(wave * 16 + srow) * LDS_STRIDE + p0];
    l1[b] = &sA[b][(wave * 16 + srow + 8) * LDS_STRIDE + p0];
  }

  v8f acc[NT];
#pragma unroll
  for (int t = 0; t < NT; ++t) acc[t] = (v8f){};

  const int C_chunks = K / KC;

  // prologue: stage chunks 0..2 into buffers 0..2 (6 async ops outstanding)
#pragma unroll
  for (int c = 0; c < NBUF - 1; ++c) {
    async_copy_b128(g0 + (size_t)c * KC, l0[c]);
    async_copy_b128(g1 + (size_t)c * KC, l1[c]);
  }

  int c = 0;
  for (; c + (NBUF - 1) < C_chunks; ++c) {
    const int pc = c + (NBUF - 1);
    // stage chunk c+3 into buf[(c+3)&3] = buf[(c-1)&3]; its previous contents
    // (chunk c-1) were fully consumed before this point in program order.
    async_copy_b128(g0 + (size_t)pc * KC, l0[pc & (NBUF - 1)]);
    async_copy_b128(g1 + (size_t)pc * KC, l1[pc & (NBUF - 1)]);
    // outstanding: chunks c..c+3 = 8 ops; wait <=6 -> chunk c landed (in-order)
    asm volatile("s_wait_asynccnt 0x6" ::: "memory");
    const float* sbase = &sA[c & (NBUF - 1)][(wave * 16 + l16) * LDS_STRIDE + kk];
    compute_chunk<NT>(sbase, Bp, c * KC, ldb, l16, half, acc);
  }
  // tail: last NBUF-1 chunks already staged; drain fully and compute.
  for (; c < C_chunks; ++c) {
    asm volatile("s_wait_asynccnt 0x0" ::: "memory");
    const float* sbase = &sA[c & (NBUF - 1)][(wave * 16 + l16) * LDS_STRIDE + kk];
    compute_chunk<NT>(sbase, Bp, c * KC, ldb, l16, half, acc);
  }

  // --- epilogue. C/D layout: VGPR v -> row m0+v (+8 for lanes 16-31),
  // col = l16 + 16t.
  if (PACKOUT) {
#pragma unroll
    for (int t = 0; t < NT; ++t) {
      const int col = t * 16 + l16;
#pragma unroll
      for (int v = 0; v < 8; v += 2) {
        const int r = m0 + v + half * 8;  // even; r+1 < M whenever r < M (M even)
        if (r < M) {
          v2f pr;
          pr.x = acc[t][v];
          pr.y = acc[t][v + 1];
          *(v2f*)(C + ((size_t)(r >> 1) * (size_t)ldb + (size_t)col) * 2) = pr;
        }
      }
    }
  } else {
#pragma unroll
    for (int t = 0; t < NT; ++t) {
      const int col = t * 16 + l16;
      float bv = 0.0f;
      if (HASBIAS) bv = bias[col];
#pragma unroll
      for (int v = 0; v < 8; ++v) {
        const int row = m0 + v + half * 8;
        if (row < M) {
          float val = acc[t][v] + bv;
          if (RELU) val = fmaxf(val, 0.0f);
          C[(size_t)row * (size_t)ldb + (size_t)col] = val;
        }
      }
    }
  }
}

// Pack B[K x N] row-major into paired-K layout:
// Bp[((k/2)*N + n)*2 + (k&1)] = B[k*N + n]
__global__ __launch_bounds__(256) void pack_pairs(const float* __restrict__ B,
                                                  float* __restrict__ Bp,
                                                  int K, int N) {
  const int i = blockIdx.x * 256 + threadIdx.x;
  if (i < K * N) {
    const int k = i / N;
    const int n = i - k * N;
    Bp[((size_t)(k >> 1) * (size_t)N + (size_t)n) * 2 + (k & 1)] = B[i];
  }
}

extern "C" void kernel_launch(void* const* d_in, const int* in_sizes, int n_in,
                              void* d_out, int out_size, void* d_ws, size_t ws_size,
                              hipStream_t stream) {
  (void)in_sizes; (void)n_in; (void)out_size; (void)ws_size;

  constexpr int N     = 10000;
  constexpr int NFEAT = 512;
  constexpr int NHID  = 256;
  constexpr int NHID2 = 128;

  const float* x   = (const float*)d_in[0];
  const float* adj = (const float*)d_in[1];
  const float* W1  = (const float*)d_in[2];
  const float* b1  = (const float*)d_in[3];
  const float* W2  = (const float*)d_in[4];
  const float* b2  = (const float*)d_in[5];
  float* out = (float*)d_out;

  // Workspace layout (~26.3 MB), all sub-offsets 16B aligned.
  float* W1p  = (float*)d_ws;                           // 512 x 256 packed
  float* W2p  = W1p  + (size_t)NFEAT * NHID;            // 256 x 128 packed
  float* XW1p = W2p  + (size_t)NHID * NHID2;            // 10000 x 256 packed
  float* H1   = XW1p + (size_t)N * NHID;                // 10000 x 256 row-major
  float* HW2p = H1   + (size_t)N * NHID;                // 10000 x 128 packed

  const dim3 blk(256);
  const dim3 grid((N + ROWS_PER_BLOCK - 1) / ROWS_PER_BLOCK);  // 79

  pack_pairs<<<dim3((NFEAT * NHID + 255) / 256), blk, 0, stream>>>(W1, W1p, NFEAT, NHID);
  pack_pairs<<<dim3((NHID * NHID2 + 255) / 256), blk, 0, stream>>>(W2, W2p, NHID, NHID2);

  // 1) XW1 = x @ W1                  (K=512)   -> packed (it is layer-2's B)
  gcn_gemm<NHID / 16, false, false, true><<<grid, blk, 0, stream>>>(
      x, W1p, nullptr, XW1p, N, NFEAT, NFEAT, NHID);

  // 2) H1 = relu(adj @ XW1 + b1)     (K=10000) [HBM-bound adj stream]
  gcn_gemm<NHID / 16, true, true, false><<<grid, blk, 0, stream>>>(
      adj, XW1p, b1, H1, N, N, N, NHID);

  // 3) HW2 = H1 @ W2                 (K=256)   -> packed (layer-4's B)
  gcn_gemm<NHID2 / 16, false, false, true><<<grid, blk, 0, stream>>>(
      H1, W2p, nullptr, HW2p, N, NHID, NHID, NHID2);

  // 4) out = relu(adj @ HW2 + b2)    (K=10000) [HBM-bound adj stream]
  gcn_gemm<NHID2 / 16, true, true, false><<<grid, blk, 0, stream>>>(
      adj, HW2p, b2, out, N, N, N, NHID2);
}